// AsymG_7155415515425
// MI455X (gfx1250) — compile-verified
//
#include <hip/hip_runtime.h>
#include <float.h>
#include <stdint.h>

// ---------------------------------------------------------------------------
// AsymG graph-attention layer for MI455X (gfx1250, wave32).
//   out = relu( seg_sum(att_pos * (h[src]Wp^T+bp), dst)
//             + seg_sum(att_neg * (h[src]Wn^T+bn), dst)
//             + h Ws^T + bs )
// GEMMs: V_WMMA_F32_16X16X4_F32 (exact fp32 matrix pipe), W staged into LDS
// via GLOBAL_LOAD_ASYNC_TO_LDS_B128 (ASYNCcnt), A reused across all 8 N-tiles,
// B operands batched per k-step so LDS latency is paid once, not 8x.
// ---------------------------------------------------------------------------

typedef float v2f __attribute__((ext_vector_type(2)));
typedef float v8f __attribute__((ext_vector_type(8)));

#define IN_DIM 128
#define EMB_DIM 64

// ---------------------------------------------------------------------------
// C[M,128] = A[M,128] @ W[128,128]^T + bias
// Block = 8 waves; whole W (64KB) async-staged to LDS once per block with a
// float4 XOR swizzle (bank-conflict-free b64 reads).  Each wave owns a 16-row
// block of A and computes all 8 column tiles (8 x v8f accumulators).
// WMMA fp32 layouts:
//   A(16x4): lane&15 = M-row, K = (lane>>4)*2 + vgpr
//   B(4x16): lane&15 = N-col, K = (lane>>4)*2 + vgpr ; B[k][n] = W[n][k]
//   D:       vgpr j -> M = j + (lane>>4)*8, N = lane&15
// ---------------------------------------------------------------------------
__global__ __launch_bounds__(256) void gemm_nt_bias_wmma(
    const float* __restrict__ A, const float* __restrict__ W,
    const float* __restrict__ bias, float* __restrict__ C, int M) {
  __shared__ float sW[128 * 128];   // 64KB, float4-XOR-swizzled rows

  // ---- cooperative async stage of W: 4096 float4s, 16 per thread ----
  for (int i = 0; i < 16; ++i) {
    int j  = (int)threadIdx.x + (i << 8);        // float4 index 0..4095
    int r  = j >> 5;                             // W row
    int c4 = j & 31;                             // float4 within row
    unsigned ldsoff = (unsigned)(uintptr_t)&sW[(r << 7) + (((c4 ^ (r & 31)) << 2))];
    unsigned long long gaddr =
        (unsigned long long)(uintptr_t)(W + ((size_t)r << 7) + ((size_t)c4 << 2));
    asm volatile("global_load_async_to_lds_b128 %0, %1, off"
                 :: "v"(ldsoff), "v"(gaddr) : "memory");
  }
  asm volatile("s_wait_asynccnt 0x0" ::: "memory");
  __syncthreads();

  const int wave  = threadIdx.x >> 5;
  const int lane  = threadIdx.x & 31;
  const int tm    = blockIdx.x * 8 + wave;       // 16-row block of output
  const int l15   = lane & 15;
  const int khalf = (lane >> 4) << 1;            // 0 or 2

  if (tm * 16 < M) {
    const float* __restrict__ arow = A + (size_t)(tm * 16 + l15) * IN_DIM + khalf;

    v8f acc[8];
#pragma unroll
    for (int tn = 0; tn < 8; ++tn) acc[tn] = (v8f){};

#pragma unroll 2
    for (int k0 = 0; k0 < IN_DIM; k0 += 4) {
      v2f a;
      a.x = arow[k0];
      a.y = arow[k0 + 1];
      const int k  = k0 + khalf;
      const int c4 = k >> 2;
      const int wi = k & 3;                      // 0 or 2: b64 stays in one float4

      // gather all 8 B operands first (distinct regs -> one dscnt wait)
      v2f b[8];
#pragma unroll
      for (int tn = 0; tn < 8; ++tn) {
        const int rowN = tn * 16 + l15;
        const int fi   = (rowN << 7) + ((c4 ^ (rowN & 31)) << 2) + wi;
        b[tn].x = sW[fi];
        b[tn].y = sW[fi + 1];
      }
      // then 8 independent WMMAs back-to-back (no D->A/B hazards)
#pragma unroll
      for (int tn = 0; tn < 8; ++tn) {
        acc[tn] = __builtin_amdgcn_wmma_f32_16x16x4_f32(false, a, false, b[tn],
                                                        (short)0, acc[tn],
                                                        false, false);
      }
    }

    const int rbase = tm * 16 + ((lane >> 4) << 3);
#pragma unroll
    for (int tn = 0; tn < 8; ++tn) {
      const int col = tn * 16 + l15;
      const float bv = bias[col];
#pragma unroll
      for (int j = 0; j < 8; ++j) {
        int r = rbase + j;
        if (r < M) C[(size_t)r * IN_DIM + col] = acc[tn][j] + bv;
      }
    }
  }
}

// ---------------------------------------------------------------------------
// Wave-per-edge raw attention score:
//   s = exp(-(||emb[d]-emb[s]|| + beta*(diff.u))) * <Q[s],K[d]> / sqrt(128)
// Also emits per-block max for the global softmax.
// ---------------------------------------------------------------------------
__global__ void edge_score_kernel(const float* __restrict__ emb,
                                  const float* __restrict__ u,
                                  const float* __restrict__ Q,
                                  const float* __restrict__ K,
                                  const int* __restrict__ src,
                                  const int* __restrict__ dst,
                                  int E, float beta,
                                  float* __restrict__ scores,
                                  float* __restrict__ blockmax) {
  const int e    = (blockIdx.x * blockDim.x + threadIdx.x) >> 5;
  const int lane = threadIdx.x & 31;
  __shared__ float smax[8];

  float score = -FLT_MAX;
  if (e < E) {
    const int s = src[e], d = dst[e];
    const float* __restrict__ es = emb + (size_t)s * EMB_DIM;
    const float* __restrict__ ed = emb + (size_t)d * EMB_DIM;
    float sq = 0.f, du = 0.f, qk = 0.f;
#pragma unroll
    for (int j = 0; j < 2; ++j) {            // 64 emb dims / 32 lanes
      int i = lane * 2 + j;
      float diff = ed[i] - es[i];
      sq += diff * diff;
      du += diff * u[i];
    }
    const float* __restrict__ q  = Q + (size_t)s * IN_DIM;
    const float* __restrict__ kk = K + (size_t)d * IN_DIM;
#pragma unroll
    for (int j = 0; j < 4; ++j) {            // 128 dims / 32 lanes
      int i = lane * 4 + j;
      qk += q[i] * kk[i];
    }
#pragma unroll
    for (int off = 16; off > 0; off >>= 1) { // full wave32 butterfly
      sq += __shfl_xor(sq, off, 32);
      du += __shfl_xor(du, off, 32);
      qk += __shfl_xor(qk, off, 32);
    }
    float dF = sqrtf(sq) + beta * du;
    score = expf(-dF) * qk * 0.08838834764831845f;   // 1/sqrt(128)
    if (lane == 0) scores[e] = score;
  }

  if (lane == 0) smax[threadIdx.x >> 5] = score;
  __syncthreads();
  if (threadIdx.x == 0) {
    float m = -FLT_MAX;
    for (int i = 0; i < (int)(blockDim.x >> 5); ++i) m = fmaxf(m, smax[i]);
    blockmax[blockIdx.x] = m;
  }
}

__global__ void reduce_max_kernel(const float* __restrict__ partial, int n,
                                  float* __restrict__ out_scalar) {
  __shared__ float sm[256];
  float m = -FLT_MAX;
  for (int i = threadIdx.x; i < n; i += blockDim.x) m = fmaxf(m, partial[i]);
  sm[threadIdx.x] = m;
  __syncthreads();
  for (int s = 128; s > 0; s >>= 1) {
    if ((int)threadIdx.x < s) sm[threadIdx.x] = fmaxf(sm[threadIdx.x], sm[threadIdx.x + s]);
    __syncthreads();
  }
  if (threadIdx.x == 0) *out_scalar = sm[0];
}

__global__ void exp_sum_kernel(const float* __restrict__ scores, int E,
                               const float* __restrict__ maxp,
                               float* __restrict__ blocksum) {
  __shared__ float sm[256];
  const int t = blockIdx.x * blockDim.x + threadIdx.x;
  const float m = *maxp;
  float v = (t < E) ? expf(scores[t] - m) : 0.f;
  sm[threadIdx.x] = v;
  __syncthreads();
  for (int s = 128; s > 0; s >>= 1) {
    if ((int)threadIdx.x < s) sm[threadIdx.x] += sm[threadIdx.x + s];
    __syncthreads();
  }
  if (threadIdx.x == 0) blocksum[blockIdx.x] = sm[0];
}

__global__ void reduce_sum_kernel(const float* __restrict__ partial, int n,
                                  float* __restrict__ out_scalar) {
  __shared__ float sm[256];
  float v = 0.f;
  for (int i = threadIdx.x; i < n; i += blockDim.x) v += partial[i];
  sm[threadIdx.x] = v;
  __syncthreads();
  for (int s = 128; s > 0; s >>= 1) {
    if ((int)threadIdx.x < s) sm[threadIdx.x] += sm[threadIdx.x + s];
    __syncthreads();
  }
  if (threadIdx.x == 0) *out_scalar = sm[0];
}

// ---------------------------------------------------------------------------
// Wave-per-edge scatter:  out[dst] += att_e * M[src]   (128 dims, float4/lane)
// att_e = exp(score_e - max) / sum
// ---------------------------------------------------------------------------
__global__ void scatter_msg_kernel(const float* __restrict__ scores,
                                   const float* __restrict__ Mbuf,
                                   const int* __restrict__ src,
                                   const int* __restrict__ dst,
                                   int E,
                                   const float* __restrict__ maxp,
                                   const float* __restrict__ sump,
                                   float* __restrict__ out) {
  const int e    = (blockIdx.x * blockDim.x + threadIdx.x) >> 5;
  const int lane = threadIdx.x & 31;
  if (e >= E) return;
  const float att = expf(scores[e] - *maxp) / *sump;
  const int s = src[e], d = dst[e];
  const float4 v = ((const float4*)(Mbuf + (size_t)s * IN_DIM))[lane];
  float* o = out + (size_t)d * IN_DIM + lane * 4;
  atomicAdd(o + 0, att * v.x);
  atomicAdd(o + 1, att * v.y);
  atomicAdd(o + 2, att * v.z);
  atomicAdd(o + 3, att * v.w);
}

__global__ void relu_kernel(float* __restrict__ out, int n) {
  int t = blockIdx.x * blockDim.x + threadIdx.x;
  if (t < n) out[t] = fmaxf(out[t], 0.f);
}

// ---------------------------------------------------------------------------
extern "C" void kernel_launch(void* const* d_in, const int* in_sizes, int n_in,
                              void* d_out, int out_size, void* d_ws, size_t ws_size,
                              hipStream_t stream) {
  const float* h     = (const float*)d_in[0];
  const float* emb   = (const float*)d_in[1];
  const float* Wq_p  = (const float*)d_in[2];
  const float* bq_p  = (const float*)d_in[3];
  const float* Wk_p  = (const float*)d_in[4];
  const float* bk_p  = (const float*)d_in[5];
  const float* u_p   = (const float*)d_in[6];
  const float* Wq_n  = (const float*)d_in[7];
  const float* bq_n  = (const float*)d_in[8];
  const float* Wk_n  = (const float*)d_in[9];
  const float* bk_n  = (const float*)d_in[10];
  const float* u_n   = (const float*)d_in[11];
  const float* W_pm  = (const float*)d_in[12];
  const float* b_pm  = (const float*)d_in[13];
  const float* W_nm  = (const float*)d_in[14];
  const float* b_nm  = (const float*)d_in[15];
  const float* W_s   = (const float*)d_in[16];
  const float* b_s   = (const float*)d_in[17];
  const int*   src_p = (const int*)d_in[18];
  const int*   dst_p = (const int*)d_in[19];
  const int*   src_n = (const int*)d_in[20];
  const int*   dst_n = (const int*)d_in[21];

  const int N = in_sizes[0] / IN_DIM;    // 50000
  const int E = in_sizes[18];            // 600000
  float* out = (float*)d_out;
  float* ws  = (float*)d_ws;

  const size_t nh = (size_t)N * IN_DIM;
  float* Qb      = ws;                   // [N,128]  (reused as M after scoring)
  float* Kb      = ws + nh;              // [N,128]
  float* scores  = ws + 2 * nh;          // [E]
  const int edgeWaveBlocks = (E + 7) / 8;        // wave/edge, 8 waves per block
  const int edgeThrdBlocks = (E + 255) / 256;    // thread/edge
  float* partial = scores + E;                   // max(edgeWaveBlocks, edgeThrdBlocks)
  float* scal    = partial + edgeWaveBlocks + 8; // [0]=max, [1]=sum

  const int tilesM     = (N + 15) / 16;
  const int gemmBlocks = (tilesM + 7) / 8;       // 8 row-tiles per block

  // self message straight into d_out (fully overwrites poison)
  gemm_nt_bias_wmma<<<gemmBlocks, 256, 0, stream>>>(h, W_s, b_s, out, N);

  struct Side {
    const float *Wq, *bq, *Wk, *bk, *u, *Wm, *bm; const int *src, *dst; float beta;
  } sides[2] = {
    { Wq_p, bq_p, Wk_p, bk_p, u_p, W_pm, b_pm, src_p, dst_p,  0.1f },
    { Wq_n, bq_n, Wk_n, bk_n, u_n, W_nm, b_nm, src_n, dst_n, -0.1f },
  };

  for (int i = 0; i < 2; ++i) {
    const Side& S = sides[i];
    gemm_nt_bias_wmma<<<gemmBlocks, 256, 0, stream>>>(h, S.Wq, S.bq, Qb, N);
    gemm_nt_bias_wmma<<<gemmBlocks, 256, 0, stream>>>(h, S.Wk, S.bk, Kb, N);
    edge_score_kernel<<<edgeWaveBlocks, 256, 0, stream>>>(emb, S.u, Qb, Kb, S.src, S.dst,
                                                          E, S.beta, scores, partial);
    reduce_max_kernel<<<1, 256, 0, stream>>>(partial, edgeWaveBlocks, scal + 0);
    exp_sum_kernel<<<edgeThrdBlocks, 256, 0, stream>>>(scores, E, scal + 0, partial);
    reduce_sum_kernel<<<1, 256, 0, stream>>>(partial, edgeThrdBlocks, scal + 1);
    // message transform reuses Q buffer
    gemm_nt_bias_wmma<<<gemmBlocks, 256, 0, stream>>>(h, S.Wm, S.bm, Qb, N);
    scatter_msg_kernel<<<edgeWaveBlocks, 256, 0, stream>>>(scores, Qb, S.src, S.dst, E,
                                                           scal + 0, scal + 1, out);
  }

  relu_kernel<<<(int)((nh + 255) / 256), 256, 0, stream>>>(out, (int)nh);
}